// AffineLog_60799557042567
// MI455X (gfx1250) — compile-verified
//
#include <hip/hip_runtime.h>

// MI455X (gfx1250) kernel: batched principal log of near-identity 4x4 affines,
// projected onto the 7-dim CSO Lie-algebra basis.
//
// Strategy (roofline: ~160 MB traffic @ 23.3 TB/s ~ 7 us; naive 24-term series
// compute would be ~5-10x that on VALU):
//  * X = A - I has a zero bottom row -> X^k = [[M^k, M^(k-1) t],[0,0]].
//  * Cayley-Hamilton on the 3x3 block M: X^4 = e1 X^3 - e2 X^2 + e3 X, so the
//    24-term Mercator series reduces EXACTLY to L = d1 X + d2 X^2 + d3 X^3,
//    with (d1,d2,d3) from a 21-step scalar recurrence. ~250 FLOPs/element.
//  * The basis projection out = Lflat(Bx12) @ W(12x7) is a real GEMM and is
//    executed with v_wmma_f32_16x16x4_f32 (3 chained K=4 steps per 16-row
//    tile; W rows 12..15 are structurally zero so the 4th step is dropped).
//    L is staged through LDS into the WMMA A-operand layout; the D-tile
//    layout gives coalesced 7-contiguous-float stores per element.

typedef __attribute__((ext_vector_type(2))) float v2f;
typedef __attribute__((ext_vector_type(4))) float v4f;
typedef __attribute__((ext_vector_type(8))) float v8f;

// Constant projection matrix W[k][n] (k = flattened L row-major index over
// rows 0..2 incl. translation column, n = output channel), evaluated per-lane
// in the WMMA B-operand layout. Pure selects -> no divergence.
__device__ __forceinline__ float bval(int k, int n) {
  const float S2 = 0.70710678118654752f;  // 1/sqrt(2)
  const float S3 = 0.57735026918962576f;  // 1/sqrt(3)
  float v = 0.0f;
  v = (k == 3  && n == 0) ?  1.0f : v;   // out0 = L03
  v = (k == 7  && n == 1) ?  1.0f : v;   // out1 = L13
  v = (k == 11 && n == 2) ?  1.0f : v;   // out2 = L23
  v = (k == 1  && n == 3) ?   S2  : v;   // out3 = (L01 - L10)/sqrt2
  v = (k == 4  && n == 3) ?  -S2  : v;
  v = (k == 2  && n == 4) ?   S2  : v;   // out4 = (L02 - L20)/sqrt2
  v = (k == 8  && n == 4) ?  -S2  : v;
  v = (k == 6  && n == 5) ?   S2  : v;   // out5 = (L12 - L21)/sqrt2
  v = (k == 9  && n == 5) ?  -S2  : v;
  v = (k == 0  && n == 6) ?   S3  : v;   // out6 = (L00+L11+L22)/sqrt3
  v = (k == 5  && n == 6) ?   S3  : v;
  v = (k == 10 && n == 6) ?   S3  : v;
  return v;
}

__global__ __launch_bounds__(256) void affine_cso_log_kernel(
    const float* __restrict__ A, float* __restrict__ out, int N) {
  // 256 rows of 12 L-entries, stride 17 floats -> conflict-free banks for
  // both the per-thread row writes and the strided A-operand gathers.
  __shared__ float Lsh[256 * 17];

  const int tid  = threadIdx.x;
  const int lane = tid & 31;
  const int wave = tid >> 5;
  const int half = lane >> 4;   // which 16-lane half of the wave
  const int col  = lane & 15;

  const int blockBase = blockIdx.x * 256;
  const int g  = blockBase + tid;
  const int ge = (g < N) ? g : (N - 1);   // clamp so EXEC stays all-ones

  // Rows 0..2 only: bottom row of A is (0,0,0,1) by construction.
  const float* ap = A + (long long)ge * 16;
  v4f r0 = *(const v4f*)(ap + 0);
  v4f r1 = *(const v4f*)(ap + 4);
  v4f r2 = *(const v4f*)(ap + 8);

  // X = A - I: M = top-left 3x3, t = top 3 of last column.
  float m00 = r0.x - 1.0f, m01 = r0.y,        m02 = r0.z,        t0 = r0.w;
  float m10 = r1.x,        m11 = r1.y - 1.0f, m12 = r1.z,        t1 = r1.w;
  float m20 = r2.x,        m21 = r2.y,        m22 = r2.z - 1.0f, t2 = r2.w;

  // Q = M^2
  float q00 = m00*m00 + m01*m10 + m02*m20;
  float q01 = m00*m01 + m01*m11 + m02*m21;
  float q02 = m00*m02 + m01*m12 + m02*m22;
  float q10 = m10*m00 + m11*m10 + m12*m20;
  float q11 = m10*m01 + m11*m11 + m12*m21;
  float q12 = m10*m02 + m11*m12 + m12*m22;
  float q20 = m20*m00 + m21*m10 + m22*m20;
  float q21 = m20*m01 + m21*m11 + m22*m21;
  float q22 = m20*m02 + m21*m12 + m22*m22;

  // C = M^3
  float c00 = q00*m00 + q01*m10 + q02*m20;
  float c01 = q00*m01 + q01*m11 + q02*m21;
  float c02 = q00*m02 + q01*m12 + q02*m22;
  float c10 = q10*m00 + q11*m10 + q12*m20;
  float c11 = q10*m01 + q11*m11 + q12*m21;
  float c12 = q10*m02 + q11*m12 + q12*m22;
  float c20 = q20*m00 + q21*m10 + q22*m20;
  float c21 = q20*m01 + q21*m11 + q22*m21;
  float c22 = q20*m02 + q21*m12 + q22*m22;

  // Translation columns of X^2, X^3: u1 = M t, u2 = M^2 t.
  float u10 = m00*t0 + m01*t1 + m02*t2;
  float u11 = m10*t0 + m11*t1 + m12*t2;
  float u12 = m20*t0 + m21*t1 + m22*t2;
  float u20 = q00*t0 + q01*t1 + q02*t2;
  float u21 = q10*t0 + q11*t1 + q12*t2;
  float u22 = q20*t0 + q21*t1 + q22*t2;

  // Newton's identities on M: X^4 = e1 X^3 - e2 X^2 + e3 X (exact, det X = 0).
  float p1 = m00 + m11 + m22;
  float p2 = q00 + q11 + q22;
  float p3 = c00 + c11 + c22;
  float e1 = p1;
  float e2 = 0.5f * (p1 * p1 - p2);
  float e3 = (e2 * p1 - e1 * p2 + p3) * (1.0f / 3.0f);
  float k3 = e1, k2 = -e2, k1 = e3;

  // Mercator series sum_{n=1}^{24} (-1)^{n+1}/n X^n reduced mod char poly:
  // L = d1 X + d2 X^2 + d3 X^3.
  float d1 = 1.0f, d2 = -0.5f, d3 = 1.0f / 3.0f;
  float b1 = 0.0f, b2 = 0.0f, b3 = 1.0f;  // coefficients of X^3
#pragma unroll
  for (int n = 4; n <= 24; ++n) {
    float nb3 = b2 + k3 * b3;
    float nb2 = b1 + k2 * b3;
    float nb1 =      k1 * b3;
    b1 = nb1; b2 = nb2; b3 = nb3;
    float an = ((n & 1) ? 1.0f : -1.0f) / (float)n;  // constant-folded
    d1 += an * b1; d2 += an * b2; d3 += an * b3;
  }

  // L rows 0..2 (row-major 3x4) -> LDS.
  float* row = &Lsh[tid * 17];
  row[0]  = d1*m00 + d2*q00 + d3*c00;
  row[1]  = d1*m01 + d2*q01 + d3*c01;
  row[2]  = d1*m02 + d2*q02 + d3*c02;
  row[3]  = d1*t0  + d2*u10 + d3*u20;
  row[4]  = d1*m10 + d2*q10 + d3*c10;
  row[5]  = d1*m11 + d2*q11 + d3*c11;
  row[6]  = d1*m12 + d2*q12 + d3*c12;
  row[7]  = d1*t1  + d2*u11 + d3*u21;
  row[8]  = d1*m20 + d2*q20 + d3*c20;
  row[9]  = d1*m21 + d2*q21 + d3*c21;
  row[10] = d1*m22 + d2*q22 + d3*c22;
  row[11] = d1*t2  + d2*u12 + d3*u22;

  __syncthreads();

  // Constant W in B-operand layout: VGPR v, lane l -> W[4s + 2*half + v][col].
  // Only K rows 0..11 are nonzero -> 3 chained WMMA steps cover the contraction.
  v2f Bv[3];
#pragma unroll
  for (int s = 0; s < 3; ++s) {
    int kb = 4 * s + 2 * half;
    Bv[s].x = bval(kb,     col);
    Bv[s].y = bval(kb + 1, col);
  }

  // Projection GEMM: out(16x7 per tile) = Lflat(16x12) @ W(12x16) via 3
  // chained v_wmma_f32_16x16x4_f32. Two tiles per wave (32 elements).
  v8f acc[2];
#pragma unroll
  for (int T = 0; T < 2; ++T) {
    const int elem = wave * 32 + T * 16 + col;  // M-row this lane feeds
    const float* lrow = &Lsh[elem * 17];
    v8f a = {};
#pragma unroll
    for (int s = 0; s < 3; ++s) {
      // A-operand layout (16x4 f32): lanes 0-15 = M, VGPR0/1 hold
      // K = 4s + 2*half (+1).
      v2f av;
      av.x = lrow[4 * s + 2 * half];
      av.y = lrow[4 * s + 2 * half + 1];
      a = __builtin_amdgcn_wmma_f32_16x16x4_f32(
          false, av, false, Bv[s], (short)0, a, false, false);
    }
    acc[T] = a;
  }

  // D layout: VGPR r -> (M = r + 8*half, N = col). Lanes 0-6 / 16-22 emit the
  // 7 contiguous outputs of two elements per store -> coalesced. Single
  // loop-invariant predicate; OOB rows hold element N-1's identical data, so
  // index-clamped duplicate stores are byte-identical and deterministic.
  if (col < 7) {
    const int mBase = blockBase + wave * 32 + 8 * half;
#pragma unroll
    for (int T = 0; T < 2; ++T) {
#pragma unroll
      for (int r = 0; r < 8; ++r) {
        int gm = mBase + T * 16 + r;
        gm = (gm < N) ? gm : (N - 1);
        out[(long long)gm * 7 + col] = acc[T][r];
      }
    }
  }
}

extern "C" void kernel_launch(void* const* d_in, const int* in_sizes, int n_in,
                              void* d_out, int out_size, void* d_ws, size_t ws_size,
                              hipStream_t stream) {
  const float* A = (const float*)d_in[0];
  float* out = (float*)d_out;
  const int N = in_sizes[0] / 16;           // number of 4x4 affines
  const int blocks = (N + 255) / 256;
  affine_cso_log_kernel<<<blocks, 256, 0, stream>>>(A, out, N);
}